// WindowAttention_mixer_82360292868144
// MI455X (gfx1250) — compile-verified
//
#include <hip/hip_runtime.h>
#include <hip/hip_bf16.h>

// ---------------------------------------------------------------------------
// WindowAttention mixer for MI455X (gfx1250), fp32 end-to-end.
// All 64x64-K GEMMs (v-proj, Q, K, QK^T, P*V, fused out+proj) run on
// V_WMMA_F32_16X16X4_F32. Memory-bound workload => fp32 matrix path is free
// and keeps numerics aligned with the reference.
// ---------------------------------------------------------------------------

#define Bn   4
#define Cc   64
#define Hh   256
#define Wdim 256
#define Pp   65536          // H*W
#define NPIX 262144         // B*P
#define NWIN 1024           // windows per batch (32x32)
#define LA   68             // LDS row stride (pad 64 -> 68: conflict-free, 8B aligned)

typedef __attribute__((ext_vector_type(2))) float v2f;
typedef __attribute__((ext_vector_type(8))) float v8f;

static __device__ __forceinline__ v8f wmma4(v2f a, v2f b, v8f c) {
  // D = A(16x4,f32) * B(4x16,f32) + C(16x16,f32)
  return __builtin_amdgcn_wmma_f32_16x16x4_f32(
      /*neg_a=*/false, a, /*neg_b=*/false, b,
      /*c_mod=*/(short)0, c, /*reuse_a=*/false, /*reuse_b=*/false);
}

static __device__ __forceinline__ float leaky(float x) { return x >= 0.f ? x : 0.1f * x; }
static __device__ __forceinline__ float sigmoidf_(float x) { return 1.f / (1.f + expf(-x)); }
static __device__ __forceinline__ float gelu_(float x) {
  return 0.5f * x * (1.f + erff(x * 0.7071067811865475f));
}

// ------------------------------- utility -----------------------------------
__global__ void k_zero(float* __restrict__ p, int n) {
  int i = blockIdx.x * blockDim.x + threadIdx.x;
  if (i < n) p[i] = 0.f;
}

// Wf = w_proj @ w_out ; bf = w_proj @ b_out + b_proj  (fuses two GEMM passes)
__global__ void k_fusew(const float* __restrict__ w_proj, const float* __restrict__ w_out,
                        const float* __restrict__ b_out, const float* __restrict__ b_proj,
                        float* __restrict__ Wf, float* __restrict__ bf) {
  int t = blockIdx.x * blockDim.x + threadIdx.x;
  if (t < 4096) {
    int o = t >> 6, c = t & 63;
    float a = 0.f;
    for (int m = 0; m < 64; ++m) a += w_proj[o * 64 + m] * w_out[m * 64 + c];
    Wf[t] = a;
  }
  if (t < 64) {
    float a = b_proj[t];
    for (int m = 0; m < 64; ++m) a += w_proj[t * 64 + m] * b_out[m];
    bf[t] = a;
  }
}

// ----------------------- generic 64->64 pixel GEMM --------------------------
// In,Out channel-major [B][64][Pp].  4 waves * 16px = 64 pixels per block.
__global__ __launch_bounds__(128) void k_gemm64(const float* __restrict__ In,
                                                const float* __restrict__ Wm,
                                                const float* __restrict__ bias,
                                                float* __restrict__ Out) {
  __shared__ float sA[64 * LA];
  __shared__ float sW[64 * LA];
  const int tid = threadIdx.x;
  const long p0 = (long)blockIdx.x * 64;
  const long b = p0 / Pp, pin = p0 % Pp;
  const float* inb = In + b * 64l * Pp + pin;
  for (int e = tid; e < 4096; e += 128) {
    int c = e >> 6, p = e & 63;
    sA[p * LA + c] = inb[(long)c * Pp + p];
  }
  for (int e = tid; e < 4096; e += 128) sW[(e >> 6) * LA + (e & 63)] = Wm[e];
  __syncthreads();
  const int l = tid & 31, wid = tid >> 5;
  const int m0 = wid * 16, lr = l & 15, kh = (l >> 4) * 2, gh = 8 * (l >> 4);
  v8f acc[4] = {};
  for (int k0 = 0; k0 < 64; k0 += 4) {
    v2f a = *(const v2f*)&sA[(m0 + lr) * LA + k0 + kh];
#pragma unroll
    for (int nt = 0; nt < 4; ++nt) {
      v2f bb = *(const v2f*)&sW[(nt * 16 + lr) * LA + k0 + kh];
      acc[nt] = wmma4(a, bb, acc[nt]);
    }
  }
  __syncthreads();  // reuse sA as output staging for coalesced channel-major store
#pragma unroll
  for (int nt = 0; nt < 4; ++nt) {
    int n = nt * 16 + lr;
    float bv = bias[n];
#pragma unroll
    for (int g = 0; g < 8; ++g) sA[n * LA + (m0 + g + gh)] = acc[nt][g] + bv;
  }
  __syncthreads();
  float* ob = Out + b * 64l * Pp + pin;
  for (int e = tid; e < 4096; e += 128) {
    int o = e >> 6, p = e & 63;
    ob[(long)o * Pp + p] = sA[o * LA + p];
  }
}

// ------------- conditioning branch: t (LN+leaky), offsets, stats ------------
__global__ __launch_bounds__(256) void k_t(const float* __restrict__ v, const float* __restrict__ cg,
                                           const float* __restrict__ w_in, const float* __restrict__ b_in,
                                           const float* __restrict__ ln_w, const float* __restrict__ ln_b,
                                           const float* __restrict__ w_off1, const float* __restrict__ b_off1,
                                           const float* __restrict__ w_off2, const float* __restrict__ b_off2,
                                           float* __restrict__ tOut, float* __restrict__ tmean,
                                           float* __restrict__ offs, float* __restrict__ tsum) {
  __shared__ float sWin[17 * 68];
  __shared__ float red[256];
  const int tid = threadIdx.x;
  for (int e = tid; e < 17 * 68; e += 256) sWin[e] = w_in[e];
  __syncthreads();
  const int p = blockIdx.x * 256 + tid;  // global pixel
  const int b = p / Pp, pin = p % Pp;
  const int hh = pin >> 8, wwp = pin & 255;
  const float* vb = v + (long)b * 64 * Pp + pin;
  float acc[17];
#pragma unroll
  for (int h = 0; h < 17; ++h) acc[h] = b_in[h];
  for (int c = 0; c < 64; ++c) {
    float x = vb[(long)c * Pp];
#pragma unroll
    for (int h = 0; h < 17; ++h) acc[h] += sWin[h * 68 + c] * x;
  }
  const float c64 = cg[((long)b * 2 + 0) * Pp + pin];
  const float c65 = cg[((long)b * 2 + 1) * Pp + pin];
  const float c66 = -1.f + (2.f / 7.f) * (float)(hh & 7);
  const float c67 = -1.f + (2.f / 7.f) * (float)(wwp & 7);
#pragma unroll
  for (int h = 0; h < 17; ++h)
    acc[h] += sWin[h * 68 + 64] * c64 + sWin[h * 68 + 65] * c65 +
              sWin[h * 68 + 66] * c66 + sWin[h * 68 + 67] * c67;
  float mu = 0.f;
#pragma unroll
  for (int h = 0; h < 17; ++h) mu += acc[h];
  mu *= (1.f / 17.f);
  float var = 0.f;
#pragma unroll
  for (int h = 0; h < 17; ++h) { float d = acc[h] - mu; var += d * d; }
  var *= (1.f / 17.f);
  const float inv = rsqrtf(var + 1e-6f);
  float tv[17], tm = 0.f;
#pragma unroll
  for (int h = 0; h < 17; ++h) {
    float u = (acc[h] - mu) * inv * ln_w[h] + ln_b[h];
    u = leaky(u);
    tv[h] = u;
    tm += u;
    tOut[((long)b * 17 + h) * Pp + pin] = u;
  }
  tmean[p] = tm * (1.f / 17.f);
  float hid[8];
#pragma unroll
  for (int i = 0; i < 8; ++i) {
    float a = b_off1[i];
#pragma unroll
    for (int h = 0; h < 17; ++h) a += w_off1[i * 17 + h] * tv[h];
    hid[i] = leaky(a);
  }
#pragma unroll
  for (int o = 0; o < 2; ++o) {
    float a = b_off2[o];
#pragma unroll
    for (int i = 0; i < 8; ++i) a += w_off2[o * 8 + i] * hid[i];
    offs[((long)b * 2 + o) * Pp + pin] = tanhf(a) * 8.f;
  }
  // per-batch channel sums for ca
  for (int h = 0; h < 17; ++h) {
    red[tid] = tv[h];
    __syncthreads();
    for (int s = 128; s > 0; s >>= 1) {
      if (tid < s) red[tid] += red[tid + s];
      __syncthreads();
    }
    if (tid == 0) atomicAdd(&tsum[b * 17 + h], red[0]);
    __syncthreads();
  }
}

__global__ void k_ca(const float* __restrict__ tsum, const float* __restrict__ w_ca,
                     const float* __restrict__ b_ca, float* __restrict__ caA) {
  int t = threadIdx.x;
  if (t < 256) {
    int b = t >> 6, o = t & 63;
    float a = b_ca[o];
    for (int h = 0; h < 17; ++h) a += w_ca[o * 17 + h] * (tsum[b * 17 + h] * (1.f / (float)Pp));
    caA[t] = sigmoidf_(a);
  }
}

__global__ __launch_bounds__(256) void k_sa(const float* __restrict__ t, const float* __restrict__ w_sa,
                                            const float* __restrict__ b_sa, float* __restrict__ sa) {
  __shared__ float sw[153];
  const int tid = threadIdx.x;
  for (int e = tid; e < 153; e += 256) sw[e] = w_sa[e];
  __syncthreads();
  const int p = blockIdx.x * 256 + tid;
  const int b = p / Pp, pin = p % Pp;
  const int y = pin >> 8, x = pin & 255;
  float a = b_sa[0];
  for (int h = 0; h < 17; ++h) {
    const float* tc = t + ((long)b * 17 + h) * Pp;
    for (int ky = 0; ky < 3; ++ky) {
      int yy = y + ky - 1;
      if (yy < 0 || yy > 255) continue;
      for (int kx = 0; kx < 3; ++kx) {
        int xx = x + kx - 1;
        if (xx < 0 || xx > 255) continue;
        a += tc[yy * 256 + xx] * sw[h * 9 + ky * 3 + kx];
      }
    }
  }
  sa[p] = sigmoidf_(a);
}

__global__ void k_mask(const float* __restrict__ tmean, const float* __restrict__ gum,
                       const float* __restrict__ w_m1, const float* __restrict__ b_m1,
                       const float* __restrict__ w_m2, const float* __restrict__ b_m2,
                       float* __restrict__ maskb) {
  int wg = blockIdx.x * blockDim.x + threadIdx.x;
  if (wg >= Bn * NWIN) return;
  int b = wg >> 10, wi = wg & 1023;
  int wy = wi >> 5, wx = wi & 31;
  const float* tb = tmean + (long)b * Pp + (wy * 8) * 256 + wx * 8;
  float hid[8];
#pragma unroll
  for (int i = 0; i < 8; ++i) hid[i] = b_m1[i];
  for (int j = 0; j < 64; ++j) {
    float tv = tb[(j >> 3) * 256 + (j & 7)];
#pragma unroll
    for (int i = 0; i < 8; ++i) hid[i] += w_m1[i * 64 + j] * tv;
  }
  float l0 = b_m2[0], l1 = b_m2[1];
#pragma unroll
  for (int i = 0; i < 8; ++i) {
    float h = leaky(hid[i]);
    l0 += w_m2[i] * h;
    l1 += w_m2[8 + i] * h;
  }
  float mx = fmaxf(l0, l1);
  float e0 = expf(l0 - mx), e1 = expf(l1 - mx);
  float p0 = e0 / (e0 + e1), p1 = e1 / (e0 + e1);
  float g0 = gum[wg * 2 + 0], g1 = gum[wg * 2 + 1];
  maskb[wg] = (p0 + g0 >= p1 + g1) ? 1.f : 0.f;  // argmax of softmax(ps+gumbel)
}

// ------------------------- flow warp: kf = xi + warp ------------------------
__global__ __launch_bounds__(256) void k_warp(const float* __restrict__ xi, const float* __restrict__ offs,
                                              float* __restrict__ kf) {
  const int p = blockIdx.x * 256 + threadIdx.x;
  const int b = p / Pp, pin = p % Pp;
  const int y = pin >> 8, x = pin & 255;
  const float fx = offs[((long)b * 2 + 0) * Pp + pin];
  const float fy = offs[((long)b * 2 + 1) * Pp + pin];
  float sx = fminf(fmaxf((float)x + fx, 0.f), 255.f);
  float sy = fminf(fmaxf((float)y + fy, 0.f), 255.f);
  float x0f = floorf(sx), y0f = floorf(sy);
  int x0 = (int)x0f, y0 = (int)y0f;
  int x1 = min(x0 + 1, 255), y1 = min(y0 + 1, 255);
  float wx_ = sx - x0f, wy_ = sy - y0f;
  float w00 = (1.f - wx_) * (1.f - wy_), w01 = wx_ * (1.f - wy_);
  float w10 = (1.f - wx_) * wy_, w11 = wx_ * wy_;
  int i00 = y0 * 256 + x0, i01 = y0 * 256 + x1, i10 = y1 * 256 + x0, i11 = y1 * 256 + x1;
  const float* xb = xi + (long)b * 64 * Pp;
  float* kb = kf + (long)b * 64 * Pp + pin;
  for (int c = 0; c < 64; ++c) {
    const float* xc = xb + (long)c * Pp;
    kb[(long)c * Pp] = xc[pin] + w00 * xc[i00] + w01 * xc[i01] + w10 * xc[i10] + w11 * xc[i11];
  }
}

// ------------------------- per-window attention -----------------------------
// One 128-thread block (4 waves) per window.  All four GEMMs on WMMA fp32.
__global__ __launch_bounds__(128) void k_attn(const float* __restrict__ xi, const float* __restrict__ kf,
                                              const float* __restrict__ vbuf, const float* __restrict__ sabuf,
                                              const float* __restrict__ maskb,
                                              const float* __restrict__ w_q, const float* __restrict__ b_q,
                                              const float* __restrict__ w_k, const float* __restrict__ b_k,
                                              const float* __restrict__ rpb, float* __restrict__ ao) {
  __shared__ float sm[5 * 64 * LA + 232];
  float* sX = sm;                 // later reused as S (scores)
  float* sKF = sm + 64 * LA;      // later reused as F staging
  float* sV = sm + 2 * 64 * LA;
  float* sQ = sm + 3 * 64 * LA;
  float* sK = sm + 4 * 64 * LA;
  float* sS = sX;
  float* sF = sKF;
  float* sRpb = sm + 5 * 64 * LA;

  const int tid = threadIdx.x;
  const int widx = blockIdx.x;
  const int b = widx >> 10, wi = widx & 1023;
  const int wy = wi >> 5, wx = wi & 31;
  const long base = (long)b * 64 * Pp;
  const int porg = (wy * 8) * 256 + wx * 8;
  const float mask = maskb[widx];

  if (mask > 0.5f) {
    for (int e = tid; e < 4096; e += 128) {
      int c = e >> 6, pos = e & 63;
      long g = base + (long)c * Pp + porg + (pos >> 3) * 256 + (pos & 7);
      sX[pos * LA + c] = xi[g];
      sKF[pos * LA + c] = kf[g];
      sV[pos * LA + c] = vbuf[g];
    }
    for (int e = tid; e < 225; e += 128) sRpb[e] = rpb[e];
    __syncthreads();

    const int l = tid & 31, wid = tid >> 5;
    const int m0 = wid * 16, lr = l & 15, kh = (l >> 4) * 2, gh = 8 * (l >> 4);

    // Q = X*Wq^T + bq ;  K = KF*Wk^T + bk
    v8f qa[4] = {}, ka[4] = {};
    for (int k0 = 0; k0 < 64; k0 += 4) {
      v2f a1 = *(const v2f*)&sX[(m0 + lr) * LA + k0 + kh];
      v2f a2 = *(const v2f*)&sKF[(m0 + lr) * LA + k0 + kh];
#pragma unroll
      for (int nt = 0; nt < 4; ++nt) {
        int n = nt * 16 + lr;
        v2f bq = *(const v2f*)&w_q[n * 64 + k0 + kh];
        v2f bk = *(const v2f*)&w_k[n * 64 + k0 + kh];
        qa[nt] = wmma4(a1, bq, qa[nt]);
        ka[nt] = wmma4(a2, bk, ka[nt]);
      }
    }
#pragma unroll
    for (int nt = 0; nt < 4; ++nt) {
      int n = nt * 16 + lr;
      float bqv = b_q[n], bkv = b_k[n];
#pragma unroll
      for (int g = 0; g < 8; ++g) {
        int m = m0 + g + gh;
        sQ[m * LA + n] = qa[nt][g] + bqv;
        sK[m * LA + n] = ka[nt][g] + bkv;
      }
    }
    __syncthreads();

    // S = Q*K^T * SCALE + rpb[relidx]
    v8f sc[4] = {};
    for (int k0 = 0; k0 < 64; k0 += 4) {
      v2f a1 = *(const v2f*)&sQ[(m0 + lr) * LA + k0 + kh];
#pragma unroll
      for (int nt = 0; nt < 4; ++nt) {
        v2f bb = *(const v2f*)&sK[(nt * 16 + lr) * LA + k0 + kh];
        sc[nt] = wmma4(a1, bb, sc[nt]);
      }
    }
#pragma unroll
    for (int nt = 0; nt < 4; ++nt) {
      int n = nt * 16 + lr;
      int jy = n >> 3, jx = n & 7;
#pragma unroll
      for (int g = 0; g < 8; ++g) {
        int m = m0 + g + gh;
        int iy = m >> 3, ix = m & 7;
        sS[m * LA + n] = sc[nt][g] * 0.125f + sRpb[(iy - jy + 7) * 15 + (ix - jx + 7)];
      }
    }
    __syncthreads();

    if (tid < 64) {  // row softmax
      float* row = &sS[tid * LA];
      float mx = row[0];
      for (int j = 1; j < 64; ++j) mx = fmaxf(mx, row[j]);
      float s = 0.f;
      for (int j = 0; j < 64; ++j) { float e = expf(row[j] - mx); row[j] = e; s += e; }
      float invs = 1.f / s;
      for (int j = 0; j < 64; ++j) row[j] *= invs;
    }
    __syncthreads();

    // F = P*V
    v8f fa[4] = {};
    for (int k0 = 0; k0 < 64; k0 += 4) {
      v2f a1 = *(const v2f*)&sS[(m0 + lr) * LA + k0 + kh];
#pragma unroll
      for (int nt = 0; nt < 4; ++nt) {
        int n = nt * 16 + lr;
        v2f bb;
        bb.x = sV[(k0 + kh) * LA + n];
        bb.y = sV[(k0 + kh + 1) * LA + n];
        fa[nt] = wmma4(a1, bb, fa[nt]);
      }
    }
#pragma unroll
    for (int nt = 0; nt < 4; ++nt) {
      int n = nt * 16 + lr;
#pragma unroll
      for (int g = 0; g < 8; ++g) sF[(m0 + g + gh) * LA + n] = fa[nt][g];
    }
    __syncthreads();
    for (int e = tid; e < 4096; e += 128) {
      int c = e >> 6, pos = e & 63;
      long g = base + (long)c * Pp + porg + (pos >> 3) * 256 + (pos & 7);
      ao[g] = sF[pos * LA + c];
    }
  } else {
    // masked-off window: out = v * sa
    for (int e = tid; e < 4096; e += 128) {
      int c = e >> 6, pos = e & 63;
      int pp = porg + (pos >> 3) * 256 + (pos & 7);
      long g = base + (long)c * Pp + pp;
      ao[g] = vbuf[g] * sabuf[(long)b * Pp + pp];
    }
  }
}

// ------------------------- depthwise 3x3 (dilated) --------------------------
__global__ __launch_bounds__(256) void k_dw(const float* __restrict__ in, const float* __restrict__ wgt,
                                            const float* __restrict__ bias, float* __restrict__ out, int dil) {
  long e = (long)blockIdx.x * 256 + threadIdx.x;  // over B*64*Pp
  int bc = (int)(e / Pp);
  int pin = (int)(e % Pp);
  int c = bc & 63;
  int y = pin >> 8, x = pin & 255;
  const float* ib = in + (long)bc * Pp;
  float a = bias[c];
  for (int ky = 0; ky < 3; ++ky) {
    int yy = y + (ky - 1) * dil;
    if (yy < 0 || yy > 255) continue;
    for (int kx = 0; kx < 3; ++kx) {
      int xx = x + (kx - 1) * dil;
      if (xx < 0 || xx > 255) continue;
      a += ib[yy * 256 + xx] * wgt[c * 9 + ky * 3 + kx];
    }
  }
  out[e] = a;
}

// ------------ final: z = gelu(conv2)*ca + ao ; out = z @ Wf^T + bf ----------
__global__ __launch_bounds__(128) void k_final(const float* __restrict__ conv2, const float* __restrict__ ao,
                                               const float* __restrict__ caA, const float* __restrict__ Wf,
                                               const float* __restrict__ bf, float* __restrict__ Out) {
  __shared__ float sA[64 * LA];
  __shared__ float sW[64 * LA];
  const int tid = threadIdx.x;
  const long p0 = (long)blockIdx.x * 64;
  const long b = p0 / Pp, pin = p0 % Pp;
  const float* c2 = conv2 + b * 64l * Pp + pin;
  const float* aop = ao + b * 64l * Pp + pin;
  const float* cab = caA + b * 64;
  for (int e = tid; e < 4096; e += 128) {
    int c = e >> 6, p = e & 63;
    float g = gelu_(c2[(long)c * Pp + p]);
    sA[p * LA + c] = g * cab[c] + aop[(long)c * Pp + p];
  }
  for (int e = tid; e < 4096; e += 128) sW[(e >> 6) * LA + (e & 63)] = Wf[e];
  __syncthreads();
  const int l = tid & 31, wid = tid >> 5;
  const int m0 = wid * 16, lr = l & 15, kh = (l >> 4) * 2, gh = 8 * (l >> 4);
  v8f acc[4] = {};
  for (int k0 = 0; k0 < 64; k0 += 4) {
    v2f a = *(const v2f*)&sA[(m0 + lr) * LA + k0 + kh];
#pragma unroll
    for (int nt = 0; nt < 4; ++nt) {
      v2f bb = *(const v2f*)&sW[(nt * 16 + lr) * LA + k0 + kh];
      acc[nt] = wmma4(a, bb, acc[nt]);
    }
  }
#pragma unroll
  for (int nt = 0; nt < 4; ++nt) {
    int n = nt * 16 + lr;
    float bv = bf[n];
#pragma unroll
    for (int g = 0; g < 8; ++g) Out[(p0 + m0 + g + gh) * 64 + n] = acc[nt][g] + bv;  // [B][P][64]
  }
}

// ---------------------------------------------------------------------------
extern "C" void kernel_launch(void* const* d_in, const int* in_sizes, int n_in,
                              void* d_out, int out_size, void* d_ws, size_t ws_size,
                              hipStream_t stream) {
  const float* x      = (const float*)d_in[0];   // (B,H*W,C) == channel-major xi when reshaped
  const float* cg     = (const float*)d_in[1];
  const float* gum    = (const float*)d_in[2];
  const float* w_v    = (const float*)d_in[3];
  const float* b_v    = (const float*)d_in[4];
  const float* w_q    = (const float*)d_in[5];
  const float* b_q    = (const float*)d_in[6];
  const float* w_k    = (const float*)d_in[7];
  const float* b_k    = (const float*)d_in[8];
  const float* w_proj = (const float*)d_in[9];
  const float* b_proj = (const float*)d_in[10];
  const float* w_out  = (const float*)d_in[11];
  const float* b_out  = (const float*)d_in[12];
  const float* rpb    = (const float*)d_in[13];
  const float* w_cs1  = (const float*)d_in[14];
  const float* b_cs1  = (const float*)d_in[15];
  const float* w_cs2  = (const float*)d_in[16];
  const float* b_cs2  = (const float*)d_in[17];
  const float* w_in   = (const float*)d_in[18];
  const float* b_in   = (const float*)d_in[19];
  const float* ln_w   = (const float*)d_in[20];
  const float* ln_b   = (const float*)d_in[21];
  const float* w_off1 = (const float*)d_in[22];
  const float* b_off1 = (const float*)d_in[23];
  const float* w_off2 = (const float*)d_in[24];
  const float* b_off2 = (const float*)d_in[25];
  const float* w_m1   = (const float*)d_in[26];
  const float* b_m1   = (const float*)d_in[27];
  const float* w_m2   = (const float*)d_in[28];
  const float* b_m2   = (const float*)d_in[29];
  const float* w_ca   = (const float*)d_in[30];
  const float* b_ca   = (const float*)d_in[31];
  const float* w_sa   = (const float*)d_in[32];
  const float* b_sa   = (const float*)d_in[33];

  // workspace layout (floats); big buffers are reused across phases
  float* w = (float*)d_ws;
  const size_t NPC = (size_t)NPIX * 64;             // 16.7M floats
  float* vbuf  = w;                                  // v        -> conv2 (reused)
  float* kfbuf = w + NPC;                            // kf       -> conv1 (reused)
  float* aobuf = w + 2 * NPC;                        // attention output (residual)
  float* tbuf  = w + 3 * NPC;                        // t: B*17*P
  float* tmeanb = tbuf + (size_t)Bn * 17 * Pp;       // NPIX
  float* sabuf  = tmeanb + NPIX;                     // NPIX
  float* offbuf = sabuf + NPIX;                      // 2*NPIX
  float* maskb  = offbuf + 2 * (size_t)NPIX;         // B*NWIN
  float* tsumb  = maskb + Bn * NWIN;                 // 68
  float* cabuf  = tsumb + 68;                        // 256
  float* Wfb    = cabuf + 256;                       // 4096
  float* bfb    = Wfb + 4096;                        // 64
  float* outp   = (float*)d_out;

  k_zero<<<1, 128, 0, stream>>>(tsumb, 68);
  k_fusew<<<16, 256, 0, stream>>>(w_proj, w_out, b_out, b_proj, Wfb, bfb);

  // v = conv1x1(xi, w_v)        (x reinterpreted as [B][64][P] channel-major)
  k_gemm64<<<NPIX / 64, 128, 0, stream>>>(x, w_v, b_v, vbuf);

  // conditioning branch
  k_t<<<NPIX / 256, 256, 0, stream>>>(vbuf, cg, w_in, b_in, ln_w, ln_b,
                                      w_off1, b_off1, w_off2, b_off2,
                                      tbuf, tmeanb, offbuf, tsumb);
  k_ca<<<1, 256, 0, stream>>>(tsumb, w_ca, b_ca, cabuf);
  k_sa<<<NPIX / 256, 256, 0, stream>>>(tbuf, w_sa, b_sa, sabuf);
  k_mask<<<(Bn * NWIN) / 256, 256, 0, stream>>>(tmeanb, gum, w_m1, b_m1, w_m2, b_m2, maskb);
  k_warp<<<NPIX / 256, 256, 0, stream>>>(x, offbuf, kfbuf);

  // window attention (WMMA) -> aobuf
  k_attn<<<Bn * NWIN, 128, 0, stream>>>(x, kfbuf, vbuf, sabuf, maskb,
                                        w_q, b_q, w_k, b_k, rpb, aobuf);

  // depthwise convs (kf region -> conv1, v region -> conv2; both dead now)
  k_dw<<<(NPIX * 64) / 256, 256, 0, stream>>>(aobuf, w_cs1, b_cs1, kfbuf, 1);
  k_dw<<<(NPIX * 64) / 256, 256, 0, stream>>>(kfbuf, w_cs2, b_cs2, vbuf, 2);

  // fused gelu*ca + residual + (w_proj@w_out) GEMM straight into d_out
  k_final<<<NPIX / 64, 128, 0, stream>>>(vbuf, aobuf, cabuf, Wfb, bfb, outp);
  (void)in_sizes; (void)n_in; (void)out_size; (void)ws_size;
}